// DenSparse_47210280518071
// MI455X (gfx1250) — compile-verified
//
#include <hip/hip_runtime.h>

#define IN_SIZE  65536
#define OUT_SIZE 65536
#define NNZ      1048576
#define BATCH    32

typedef __attribute__((ext_vector_type(2))) float v2f;
typedef __attribute__((ext_vector_type(4))) float v4f;
typedef __attribute__((ext_vector_type(8))) float v8f;

// ---------------------------------------------------------------------------
// Zero a float buffer (float4 grid-stride).
// ---------------------------------------------------------------------------
__global__ void zero_f4_kernel(float4* __restrict__ p, int n4) {
    int i = blockIdx.x * blockDim.x + threadIdx.x;
    int stride = gridDim.x * blockDim.x;
    for (; i < n4; i += stride) p[i] = make_float4(0.f, 0.f, 0.f, 0.f);
}

// ---------------------------------------------------------------------------
// x [32, IN] -> xT [IN, 32] via padded LDS tile (coalesced both sides).
// Block (32, 8); one 32x32 tile per block.
// ---------------------------------------------------------------------------
__global__ void xpose_x_kernel(const float* __restrict__ x, float* __restrict__ xT) {
    __shared__ float tile[32][33];
    const int tx = threadIdx.x;
    const int o  = blockIdx.x * 32 + tx;
#pragma unroll
    for (int i = threadIdx.y; i < 32; i += 8)
        tile[i][tx] = x[(size_t)i * IN_SIZE + o];
    __syncthreads();
#pragma unroll
    for (int i = threadIdx.y; i < 32; i += 8)
        xT[((size_t)blockIdx.x * 32 + i) * BATCH + tx] = tile[tx][i];
}

// ---------------------------------------------------------------------------
// Edge scatter: one wave32 per edge, lane = batch index.
// Gather is one 128B line (xT row), scatter is one coalesced 128B atomic line.
// ---------------------------------------------------------------------------
__global__ void edge_kernel(const float* __restrict__ xT,
                            const float* __restrict__ w,
                            const int*   __restrict__ dst,
                            const int*   __restrict__ src,
                            float*       __restrict__ outT, int nnz) {
    const int lane   = threadIdx.x & 31;
    const int wave   = (blockIdx.x * blockDim.x + threadIdx.x) >> 5;
    const int nwaves = (gridDim.x * blockDim.x) >> 5;
    for (int e = wave; e < nnz; e += nwaves) {
        // stream next iteration's metadata through L2 (global_prefetch_b8)
        if (e + nwaves < nnz) {
            __builtin_prefetch(&w[e + nwaves],   0, 0);
            __builtin_prefetch(&src[e + nwaves], 0, 0);
            __builtin_prefetch(&dst[e + nwaves], 0, 0);
        }
        const float we = w[e];      // uniform across wave -> single request
        const int   s  = src[e];
        const int   d  = dst[e];
        const float xv = xT[(size_t)s * BATCH + lane];
        atomicAdd(&outT[(size_t)d * BATCH + lane], we * xv);
    }
}

// ---------------------------------------------------------------------------
// outT [OUT, 32] -> out [32, OUT] using v_wmma_f32_16x16x4_f32.
//
// A-layout (16x4 f32): lane = m + 16*(k>>1), vgpr = k&1  (documented)
// C/D layout (16x16):  lane = n + 16*(m>>3), vgpr = m&7  (documented)
// With B_j[k][n] = (n == k + 4j), D = sum_j A_j x B_j gives
//   D[m][n] = inTile[o = obase+m][b = g*16 + n]
// i.e. data migrates from lane~o (A) to lane~b (D): a cross-lane transpose.
// One wave handles a 16-o tile (8 wmmas: 2 batch halves x 4 chained).
// ---------------------------------------------------------------------------
__global__ void wmma_xpose_out_kernel(const float* __restrict__ outT,
                                      float* __restrict__ out) {
    const int lane  = threadIdx.x & 31;
    const int wave  = blockIdx.x * (blockDim.x >> 5) + (threadIdx.x >> 5);
    const int obase = wave * 16;                 // 4096 tiles total, exact fit
    const int r     = lane & 15;                 // row within tile (A) / col n (D)
    const int h     = lane >> 4;                 // half-wave selector

    const float* rowp = outT + (size_t)(obase + r) * BATCH;

#pragma unroll
    for (int g = 0; g < 2; ++g) {                // batch halves b=0..15, b=16..31
        v8f d = {};                              // zero accumulator
#pragma unroll
        for (int j = 0; j < 4; ++j) {
            const int col = g * 16 + 4 * j + 2 * h;
            v2f a = *(const v2f*)(rowp + col);   // A_j fragment (b64 load)
            v2f b;                               // B_j = shifted 4x16 identity
            b.x = (r == (2 * h + 4 * j    )) ? 1.0f : 0.0f;
            b.y = (r == (2 * h + 4 * j + 1)) ? 1.0f : 0.0f;
            d = __builtin_amdgcn_wmma_f32_16x16x4_f32(
                    /*neg_a=*/false, a, /*neg_b=*/false, b,
                    /*c_mod=*/(short)0, d, /*reuse_a=*/false, /*reuse_b=*/false);
        }
        // lane holds out[b = g*16 + r][o = obase + 8h + 0..7], contiguous in o
        float* op = out + (size_t)(g * 16 + r) * OUT_SIZE + obase + 8 * h;
        v4f lo = { d[0], d[1], d[2], d[3] };
        v4f hi = { d[4], d[5], d[6], d[7] };
        *(v4f*)(op)     = lo;
        *(v4f*)(op + 4) = hi;
    }
}

// ---------------------------------------------------------------------------
// Fallback (ws too small): direct atomic scatter into d_out.
// ---------------------------------------------------------------------------
__global__ void edge_naive_kernel(const float* __restrict__ x,
                                  const float* __restrict__ w,
                                  const int*   __restrict__ dst,
                                  const int*   __restrict__ src,
                                  float*       __restrict__ out, int nnz) {
    const int lane   = threadIdx.x & 31;
    const int wave   = (blockIdx.x * blockDim.x + threadIdx.x) >> 5;
    const int nwaves = (gridDim.x * blockDim.x) >> 5;
    for (int e = wave; e < nnz; e += nwaves) {
        const float we = w[e];
        const int   s  = src[e];
        const int   d  = dst[e];
        atomicAdd(&out[(size_t)lane * OUT_SIZE + d],
                  we * x[(size_t)lane * IN_SIZE + s]);
    }
}

// ---------------------------------------------------------------------------
extern "C" void kernel_launch(void* const* d_in, const int* in_sizes, int n_in,
                              void* d_out, int out_size, void* d_ws, size_t ws_size,
                              hipStream_t stream) {
    const float* x   = (const float*)d_in[0];  // [32, 65536]
    const float* w   = (const float*)d_in[1];  // [NNZ]
    const int*   dst = (const int*)  d_in[2];  // [NNZ]
    const int*   src = (const int*)  d_in[3];  // [NNZ]
    float*       out = (float*)d_out;          // [32, 65536]

    const size_t xT_elems   = (size_t)IN_SIZE * BATCH;
    const size_t outT_elems = (size_t)OUT_SIZE * BATCH;
    const size_t need = (xT_elems + outT_elems) * sizeof(float);

    if (ws_size >= need) {
        float* xT   = (float*)d_ws;
        float* outT = xT + xT_elems;

        zero_f4_kernel<<<2048, 256, 0, stream>>>((float4*)outT,
                                                 (int)(outT_elems / 4));
        xpose_x_kernel<<<IN_SIZE / 32, dim3(32, 8), 0, stream>>>(x, xT);
        edge_kernel<<<4096, 256, 0, stream>>>(xT, w, dst, src, outT, NNZ);
        // 4096 o-tiles, 8 waves/block -> 512 blocks
        wmma_xpose_out_kernel<<<OUT_SIZE / 16 / 8, 256, 0, stream>>>(outT, out);
    } else {
        zero_f4_kernel<<<2048, 256, 0, stream>>>((float4*)out,
                                                 (int)(outT_elems / 4));
        edge_naive_kernel<<<4096, 256, 0, stream>>>(x, w, dst, src, out, NNZ);
    }
}